// MultiPositionTransfer_38663295598666
// MI455X (gfx1250) — compile-verified
//
#include <hip/hip_runtime.h>
#include <math.h>

typedef __attribute__((ext_vector_type(2))) float v2f;
typedef __attribute__((ext_vector_type(8))) float v8f;

#define NTOK 1024      // T*B = 64*16
#define DD   512       // feature dim
#define NP   17        // position buckets
#define MAXT32 48      // worst case sum ceil(n_p/32) <= 48

// ---------------------------------------------------------------------------
// Kernel 1: bucketize tokens by clamped position, build sorted index + 32-row
// tiles (tileB = bucket, tileOff = start in sorted[], tileValid = rows used)
// ---------------------------------------------------------------------------
__global__ void __launch_bounds__(1024)
bucket_kernel(const int* __restrict__ positions,
              int* __restrict__ sorted, int* __restrict__ tileB,
              int* __restrict__ tileOff, int* __restrict__ tileValid,
              int* __restrict__ numTiles)
{
    __shared__ int cnt[NP], cursor[NP], off[NP];
    int tid = threadIdx.x;
    if (tid < NP) cnt[tid] = 0;
    __syncthreads();
    int p = positions[tid];
    p = (p < 16) ? p : 16;
    atomicAdd(&cnt[p], 1);
    __syncthreads();
    if (tid == 0) {
        int o = 0, nt = 0;
        for (int q = 0; q < NP; ++q) {
            off[q] = o;
            int n = cnt[q];
            int ntl = (n + 31) >> 5;
            for (int i = 0; i < ntl; ++i) {
                tileB[nt]     = q;
                tileOff[nt]   = o + i * 32;
                int v = n - i * 32;
                tileValid[nt] = (v < 32) ? v : 32;
                ++nt;
            }
            o += n;
        }
        *numTiles = nt;
    }
    __syncthreads();
    if (tid < NP) cursor[tid] = off[tid];
    __syncthreads();
    int r = atomicAdd(&cursor[p], 1);
    sorted[r] = tid;
}

// ---------------------------------------------------------------------------
// Kernel 2: grouped GEMM via V_WMMA_F32_16X16X4_F32
// Block = 128 threads (4 waves) computes M=32 tokens x N=64 cols, one feature.
// K staged in 32-deep chunks, double-buffered in LDS:
//   As[2][32][36]  (pad 36: 36*r mod 64 distinct for r=0..15 -> conflict-free)
//   Bs[2][32][72]  (pad 72: +2 rows = bank offset 16 -> lane halves disjoint)
// Each wave holds two v8f accumulators (row sets 0-15 / 16-31) reusing each
// B fragment twice -> half the L2 weight traffic of a 16-row tile.
// ---------------------------------------------------------------------------
__global__ void __launch_bounds__(128)
gemm_kernel(const float* __restrict__ X, const float* __restrict__ Wl,
            const float* __restrict__ bl, const int* __restrict__ sorted,
            const int* __restrict__ tileB, const int* __restrict__ tileOff,
            const int* __restrict__ tileValid, const int* __restrict__ numTiles,
            float* __restrict__ S)
{
    int tile = blockIdx.x;
    if (tile >= *numTiles) return;   // block-uniform: EXEC all-1s for WMMA

    __shared__ float As[2][32][36];
    __shared__ float Bs[2][32][72];
    __shared__ int   tok[32];

    const int tid   = threadIdx.x;
    const int lane  = tid & 31;
    const int wv    = tid >> 5;
    const int f     = blockIdx.y >> 3;
    const int nbase = (blockIdx.y & 7) << 6;      // block col base in [0,512)
    const int wcol  = (wv << 4) | (lane & 15);    // col within 64-wide panel
    const int mrow  = lane & 15;
    const int koff  = (lane >> 4) << 1;           // {0,2} per lane half

    const int p     = tileB[tile];
    const int toff  = tileOff[tile];
    const int valid = tileValid[tile];

    if (tid < 32) {
        int idx = (tid < valid) ? tid : (valid - 1);  // pad rows duplicate last
        tok[tid] = sorted[toff + idx];
    }
    __syncthreads();

    const float* Wfp = Wl + ((size_t)f * NP + p) * DD * DD + nbase;

    auto stage = [&](int c, int b) {
        // B chunk: 32 K-rows x 64 cols = 512 float4, 4 per thread, coalesced
        const float* wsrc = Wfp + (size_t)(c * 32) * DD;
#pragma unroll
        for (int j = 0; j < 4; ++j) {
            int i = tid + j * 128;
            int r = i >> 4, c4 = i & 15;
            float4 v = ((const float4*)(wsrc + (size_t)r * DD))[c4];
            *(float4*)&Bs[b][r][c4 * 4] = v;
        }
        // A chunk: 32 token-rows x 32 K = 256 float4, 2 per thread
#pragma unroll
        for (int j = 0; j < 2; ++j) {
            int i = tid + j * 128;
            int r = i >> 3, c4 = i & 7;
            float4 v = ((const float4*)(X + (size_t)tok[r] * DD + c * 32))[c4];
            *(float4*)&As[b][r][c4 * 4] = v;
        }
    };

    v8f acc0 = {}, acc1 = {};

    stage(0, 0);
    __syncthreads();
    for (int c = 0; c < 16; ++c) {
        if (c + 1 < 16) stage(c + 1, (c + 1) & 1);
        const float (*A)[36] = As[c & 1];
        const float (*B)[72] = Bs[c & 1];
#pragma unroll
        for (int ks = 0; ks < 8; ++ks) {
            int kk = ks * 4 + koff;
            v2f a0, a1, b;
            a0.x = A[mrow][kk];        a0.y = A[mrow][kk + 1];
            a1.x = A[16 + mrow][kk];   a1.y = A[16 + mrow][kk + 1];
            b.x  = B[kk][wcol];        b.y  = B[kk + 1][wcol];
            acc0 = __builtin_amdgcn_wmma_f32_16x16x4_f32(
                false, a0, false, b, (short)0, acc0, false, false);
            acc1 = __builtin_amdgcn_wmma_f32_16x16x4_f32(
                false, a1, false, b, (short)0, acc1, false, false);
        }
        __syncthreads();
    }

    // C layout: vgpr v -> row = v + (lane>>4)*8, col = lane&15. Bias + scatter.
    const float bias  = bl[((size_t)f * NP + p) * DD + nbase + wcol];
    const int   rbase = (lane >> 4) << 3;
    float* Sf = S + (size_t)f * NTOK * DD + nbase + wcol;
#pragma unroll
    for (int v = 0; v < 8; ++v) {
        int r0 = rbase + v;
        if (r0 < valid) Sf[(size_t)tok[r0] * DD] = acc0[v] + bias;
        int r1 = 16 + rbase + v;
        if (r1 < valid) Sf[(size_t)tok[r1] * DD] = acc1[v] + bias;
    }
}

// ---------------------------------------------------------------------------
// Kernel 3: gating elementwise combine (float4 vectorized)
// ---------------------------------------------------------------------------
__device__ __forceinline__ float sigm(float v) { return 1.0f / (1.0f + __expf(-v)); }
__device__ __forceinline__ float gate1(float x, float s0, float s1, float s2, float s3) {
    float nr = x * sigm(s0) + tanhf(s1) * sigm(s2);
    return tanhf(nr) * sigm(s3);
}

__global__ void __launch_bounds__(256)
gate_kernel(const float* __restrict__ xin, const float* __restrict__ S,
            float* __restrict__ xout)
{
    int i = blockIdx.x * blockDim.x + threadIdx.x;   // float4 index
    const int N4 = (NTOK * DD) / 4;
    float4 s0 = ((const float4*)S)[0 * N4 + i];
    float4 s1 = ((const float4*)S)[1 * N4 + i];
    float4 s2 = ((const float4*)S)[2 * N4 + i];
    float4 s3 = ((const float4*)S)[3 * N4 + i];
    float4 x  = ((const float4*)xin)[i];
    float4 o;
    o.x = gate1(x.x, s0.x, s1.x, s2.x, s3.x);
    o.y = gate1(x.y, s0.y, s1.y, s2.y, s3.y);
    o.z = gate1(x.z, s0.z, s1.z, s2.z, s3.z);
    o.w = gate1(x.w, s0.w, s1.w, s2.w, s3.w);
    ((float4*)xout)[i] = o;
}

// ---------------------------------------------------------------------------
extern "C" void kernel_launch(void* const* d_in, const int* in_sizes, int n_in,
                              void* d_out, int out_size, void* d_ws, size_t ws_size,
                              hipStream_t stream) {
    const int*   positions = (const int*)d_in[0];    // [T,B] flat = token id
    const float* outputs   = (const float*)d_in[1];  // [1024, 512]
    const float* Ws        = (const float*)d_in[2];  // [L,4,17,512,512]
    const float* bs        = (const float*)d_in[3];  // [L,4,17,512]
    float*       out       = (float*)d_out;

    char* ws        = (char*)d_ws;
    int*  sorted    = (int*)(ws);
    int*  tileB     = (int*)(ws + 4096);
    int*  tileOff   = (int*)(ws + 4608);
    int*  tileValid = (int*)(ws + 5120);
    int*  numTiles  = (int*)(ws + 5632);
    float* S        = (float*)(ws + 8192);                                         // 8 MB
    float* xmid     = (float*)(ws + 8192 + (size_t)4 * NTOK * DD * sizeof(float)); // 2 MB

    bucket_kernel<<<1, 1024, 0, stream>>>(positions, sorted, tileB, tileOff,
                                          tileValid, numTiles);

    for (int l = 0; l < 2; ++l) {
        const float* x_in  = (l == 0) ? outputs : xmid;
        float*       x_out = (l == 1) ? out : xmid;
        const float* Wl = Ws + (size_t)l * 4 * NP * DD * DD;
        const float* bl = bs + (size_t)l * 4 * NP * DD;
        gemm_kernel<<<dim3(MAXT32, 32), 128, 0, stream>>>(
            x_in, Wl, bl, sorted, tileB, tileOff, tileValid, numTiles, S);
        gate_kernel<<<dim3((NTOK * DD) / 4 / 256), 256, 0, stream>>>(x_in, S, x_out);
    }
}